// IFFT2_14809047237124
// MI455X (gfx1250) — compile-verified
//
#include <hip/hip_runtime.h>
#include <math.h>

typedef float v2f __attribute__((ext_vector_type(2)));
typedef float v8f __attribute__((ext_vector_type(8)));
typedef float vf4 __attribute__((ext_vector_type(4)));

#define WAVES_PER_BLOCK 8
#define IPW 4                 // images per wave
#define PI_F 3.14159265358979323846f

// One wave = one image (b,c). Out(64x64) = U(64x40) * V(40x20) * Kc(20x64),
// all via V_WMMA_F32_16X16X4_F32 (exact f32, problem is HBM-write bound).
__global__ __launch_bounds__(256) void ifft2_sparse_wmma(
    const float* __restrict__ x, float* __restrict__ out, int n_images)
{
    __shared__ float sU[64 * 40];                 // [Ec | Es] / 8
    __shared__ float sK[20 * 64];                 // [C ; S]
    __shared__ float sIn[WAVES_PER_BLOCK][400];   // raw input per image
    __shared__ float sG[WAVES_PER_BLOCK][16 * 20];// [Gr|Gi] y-tile interchange

    const int tid  = threadIdx.x;
    const int wave = tid >> 5;
    const int lane = tid & 31;
    const int l15  = lane & 15;
    const int hi   = lane >> 4;    // 0: lanes 0-15, 1: lanes 16-31

    // ---- build constant matrices (amortized over 32 images/block) ----
    for (int i = tid; i < 64 * 40; i += 256) {
        int y = i / 40, j = i % 40;
        int jm = j % 20;
        int f  = (jm < 10) ? jm : (44 + jm);            // ky: 0..9 or 54..63
        float ang = (2.0f * PI_F / 64.0f) * (float)(f * y);
        sU[i] = ((j < 20) ? cosf(ang) : sinf(ang)) * 0.125f;
    }
    for (int i = tid; i < 20 * 64; i += 256) {
        int k = i / 64, xx = i % 64;
        float v;
        if (k == 0)       v = 0.125f;                                   // DC
        else if (k < 10)  v =  0.25f * cosf((2.0f*PI_F/64.0f)*(float)(k*xx));
        else if (k == 10) v = 0.0f;                                     // S row k=0
        else              v = -0.25f * sinf((2.0f*PI_F/64.0f)*(float)((k-10)*xx));
        sK[i] = v;
    }
    __syncthreads();

    for (int it = 0; it < IPW; ++it) {
        long img = ((long)blockIdx.x * WAVES_PER_BLOCK + wave) * IPW + it;
        bool active = img < n_images;   // wave-uniform -> EXEC all-ones inside

        // ---- stage raw input (400 floats) into LDS, coalesced NT loads ----
        if (active) {
            const vf4* src = (const vf4*)(x + img * 400);
            vf4* dst = (vf4*)(&sIn[wave][0]);
            for (int i = lane; i < 100; i += 32)
                dst[i] = __builtin_nontemporal_load(src + i);
        }
        __syncthreads();

        // ---- gather stage-1 B operand V(40x20) into B-tile register layout ----
        // V = [[Br, Bi], [-Bi, Br]],  Br/Bi from x[blk, a, k, {0,1}], j = blk*10+a
        v2f vB[10][2];
        if (active) {
            for (int kk = 0; kk < 10; ++kk) {
                for (int nt = 0; nt < 2; ++nt) {
                    float vv[2];
                    for (int r = 0; r < 2; ++r) {
                        int jp = 4 * kk + 2 * hi + r;   // K-row of V
                        int n  = 16 * nt + l15;         // column of V
                        float v = 0.0f;
                        if (n < 20) {
                            int jm = (jp < 20) ? jp : jp - 20;
                            int blk = jm / 10, a = jm % 10;
                            int k, ri; float sgn = 1.0f;
                            if (n < 10) { k = n;      ri = (jp < 20) ? 0 : 1;
                                          if (jp >= 20) sgn = -1.0f; }
                            else        { k = n - 10; ri = (jp < 20) ? 1 : 0; }
                            v = sgn * sIn[wave][(blk * 100 + a * 10 + k) * 2 + ri];
                        }
                        vv[r] = v;
                    }
                    v2f b; b.x = vv[0]; b.y = vv[1];
                    vB[kk][nt] = b;
                }
            }
        }

        // ---- per 16-row y-tile: stage1 WMMA -> LDS interchange -> stage2 WMMA ----
        for (int t = 0; t < 4; ++t) {
            if (active) {
                v8f d0 = {}, d1 = {};
                int y = 16 * t + l15;
                #pragma unroll
                for (int kk = 0; kk < 10; ++kk) {
                    v2f a;
                    a.x = sU[y * 40 + 4 * kk + 2 * hi + 0];
                    a.y = sU[y * 40 + 4 * kk + 2 * hi + 1];
                    d0 = __builtin_amdgcn_wmma_f32_16x16x4_f32(
                            false, a, false, vB[kk][0], (short)0, d0, false, false);
                    d1 = __builtin_amdgcn_wmma_f32_16x16x4_f32(
                            false, a, false, vB[kk][1], (short)0, d1, false, false);
                }
                // scatter D (C/D layout) into sG[16][20] for A-operand reload
                #pragma unroll
                for (int r = 0; r < 8; ++r) {
                    int M = 8 * hi + r;
                    sG[wave][M * 20 + l15] = d0[r];
                    if (l15 < 4) sG[wave][M * 20 + 16 + l15] = d1[r];
                }
            }
            __syncthreads();

            if (active) {
                v8f c0 = {}, c1 = {}, c2 = {}, c3 = {};
                #pragma unroll
                for (int kk = 0; kk < 5; ++kk) {
                    v2f a;
                    a.x = sG[wave][l15 * 20 + 4 * kk + 2 * hi + 0];
                    a.y = sG[wave][l15 * 20 + 4 * kk + 2 * hi + 1];
                    v2f b0, b1, b2, b3;
                    int r0 = (4 * kk + 2 * hi + 0) * 64;
                    int r1 = (4 * kk + 2 * hi + 1) * 64;
                    b0.x = sK[r0 +  0 + l15]; b0.y = sK[r1 +  0 + l15];
                    b1.x = sK[r0 + 16 + l15]; b1.y = sK[r1 + 16 + l15];
                    b2.x = sK[r0 + 32 + l15]; b2.y = sK[r1 + 32 + l15];
                    b3.x = sK[r0 + 48 + l15]; b3.y = sK[r1 + 48 + l15];
                    c0 = __builtin_amdgcn_wmma_f32_16x16x4_f32(false, a, false, b0, (short)0, c0, false, false);
                    c1 = __builtin_amdgcn_wmma_f32_16x16x4_f32(false, a, false, b1, (short)0, c1, false, false);
                    c2 = __builtin_amdgcn_wmma_f32_16x16x4_f32(false, a, false, b2, (short)0, c2, false, false);
                    c3 = __builtin_amdgcn_wmma_f32_16x16x4_f32(false, a, false, b3, (short)0, c3, false, false);
                }
                // store 16x64 output rows, non-temporal (write-once stream)
                float* o = out + img * 4096;
                #pragma unroll
                for (int r = 0; r < 8; ++r) {
                    int yrow = 16 * t + 8 * hi + r;
                    float* orow = o + yrow * 64 + l15;
                    __builtin_nontemporal_store(c0[r], orow +  0);
                    __builtin_nontemporal_store(c1[r], orow + 16);
                    __builtin_nontemporal_store(c2[r], orow + 32);
                    __builtin_nontemporal_store(c3[r], orow + 48);
                }
            }
            __syncthreads();
        }
    }
}

extern "C" void kernel_launch(void* const* d_in, const int* in_sizes, int n_in,
                              void* d_out, int out_size, void* d_ws, size_t ws_size,
                              hipStream_t stream) {
    const float* x = (const float*)d_in[0];
    float* out = (float*)d_out;
    long total = in_sizes[0];               // B * 2 * 2 * 100 * 2
    int n_images = (int)(total / 400);      // B * 2
    int imgs_per_block = WAVES_PER_BLOCK * IPW;
    int blocks = (n_images + imgs_per_block - 1) / imgs_per_block;
    ifft2_sparse_wmma<<<blocks, 256, 0, stream>>>(x, out, n_images);
}